// SelfAttention_9594956939834
// MI455X (gfx1250) — compile-verified
//
#include <hip/hip_runtime.h>
#include <hip/hip_bf16.h>

// Problem constants (from reference): B=2, S=2048, E=1024, H=16, D=64
static constexpr int BB = 2;
static constexpr int SS = 2048;
static constexpr int EE = 1024;
static constexpr int APAD = 16;   // zero-padded A rows for the 16x16x4 WMMA

typedef __attribute__((ext_vector_type(2))) float v2f;
typedef __attribute__((ext_vector_type(8))) float v8f;

// ---------------------------------------------------------------------------
// Kernel 1: colsum_pad[b][e] = sum_s values[b][s][e]   (rows 2..15 stay zero)
// grid (E/256, S/32, B), block 256. Coalesced 1KB row segments per iteration;
// partials combined with GLOBAL_ATOMIC_ADD_F32.
// ---------------------------------------------------------------------------
__global__ void __launch_bounds__(256)
colsum_kernel(const float* __restrict__ values, float* __restrict__ colsum_pad) {
    const int e  = blockIdx.x * 256 + threadIdx.x;
    const int s0 = blockIdx.y * 32;
    const int b  = blockIdx.z;
    const float* p = values + ((size_t)b * SS + s0) * EE + e;
    float acc = 0.0f;
#pragma unroll 8
    for (int s = 0; s < 32; ++s) {
        acc += p[(size_t)s * EE];
    }
    atomicAdd(&colsum_pad[b * EE + e], acc);
}

// ---------------------------------------------------------------------------
// Kernel 2: hsum[b][n] = colsum[b][:] . Wv[n][:] + S * bv[n]
// One wave per 16-wide N tile (64 tiles). Full-f32 matrix op:
//   V_WMMA_F32_16X16X4_F32, K-loop over E in steps of 4.
// A (16x4 f32, ISA 7.12.2): lane m=lane&15 is row M; lanes<16 hold K={k,k+1},
//   lanes>=16 hold K={k+2,k+3}. A is read from the 16-row zero-padded staging
//   buffer, so loads are unconditional b64 (no exec-mask branching).
// B (4x16 f32): B[k][n] = Wv[n][k]; mirrored lane-half K split -> b64 loads.
// C (16x16 f32): VGPR r = row M (lanes<16: M=r), lane&15 = N. Rows 0,1 stored.
// ---------------------------------------------------------------------------
__global__ void __launch_bounds__(32)
vproj_wmma_kernel(const float* __restrict__ colsum_pad,
                  const float* __restrict__ Wv,
                  const float* __restrict__ bv,
                  float* __restrict__ hsum) {
    const int tile = blockIdx.x;            // n-tile: columns [tile*16, tile*16+16)
    const int lane = threadIdx.x;           // 0..31
    const int half = lane >> 4;             // 0 -> K {k,k+1}, 1 -> K {k+2,k+3}
    const int m    = lane & 15;             // A row M (batch index; rows >=2 are zero)
    const int n    = tile * 16 + (lane & 15);

    const float* arow = colsum_pad + (size_t)m * EE + half * 2;
    const float* wrow = Wv + (size_t)n * EE + half * 2;  // B[k][n] = Wv[n][k]

    v8f c = {0.f, 0.f, 0.f, 0.f, 0.f, 0.f, 0.f, 0.f};

#pragma unroll 4
    for (int k = 0; k < EE; k += 4) {
        const v2f a  = *(const v2f*)(arow + k);   // unconditional b64
        const v2f bm = *(const v2f*)(wrow + k);   // unconditional b64
        // (neg_a, A, neg_b, B, c_mod, C, reuse_a, reuse_b)
        c = __builtin_amdgcn_wmma_f32_16x16x4_f32(
                false, a, false, bm, (short)0, c, false, false);
    }

    if (half == 0) {
        const float bias = (float)SS * bv[n];
        hsum[0 * EE + n] = c[0] + bias;   // M=0 lives in VGPR 0, lanes 0..15
        hsum[1 * EE + n] = c[1] + bias;   // M=1 lives in VGPR 1, lanes 0..15
    }
}

// ---------------------------------------------------------------------------
// Kernel 3: out[b][s][e] = hsum[b][e]  (broadcast over s), float4 stores.
// out flat index ((b*S+s)*H+h)*D+d == (b*S+s)*E + (h*D+d), so this is exact.
// ---------------------------------------------------------------------------
__global__ void __launch_bounds__(256)
bcast_kernel(const float* __restrict__ hsum, float* __restrict__ out) {
    const size_t i  = (size_t)blockIdx.x * 256 + threadIdx.x;  // over B*S*(E/4)
    const int    e4 = (int)(i & (EE / 4 - 1));
    const size_t bs = i >> 8;                                   // E/4 == 256
    const int    b  = (bs >= (size_t)SS) ? 1 : 0;
    const float4 v  = ((const float4*)(hsum + (size_t)b * EE))[e4];
    ((float4*)out)[i] = v;
}

// ---------------------------------------------------------------------------
// Launcher. Inputs: 0=values 1=keys 2=queries 3=Wv 4=bv 5=Wk 6=bk 7=Wq 8=bq.
// Reference einsum 'bhqk,bvhd->bqhd' factorizes as
//   (sum_k softmax(energy)) * (sum_v v)  ==  1 * (sum_v v),
// so the Q/K/softmax pipeline is mathematically dead code; by linearity
//   out[b,q,h,d] = (sum_s values[b,s,:]) @ Wv[h*D+d,:] + S * bv[h*D+d].
// ---------------------------------------------------------------------------
extern "C" void kernel_launch(void* const* d_in, const int* in_sizes, int n_in,
                              void* d_out, int out_size, void* d_ws, size_t ws_size,
                              hipStream_t stream) {
    (void)in_sizes; (void)n_in; (void)out_size; (void)ws_size;

    const float* values = (const float*)d_in[0];
    const float* Wv     = (const float*)d_in[3];
    const float* bv     = (const float*)d_in[4];
    float*       out    = (float*)d_out;

    float* colsum_pad = (float*)d_ws;                   // 16*E f32 = 64 KB
    float* hsum       = (float*)d_ws + APAD * EE;       //  2*E f32 =  8 KB

    // A staging buffer accumulates via atomics and relies on rows 2..15 being
    // zero -> must be cleared every launch (memset node is graph-capturable).
    hipMemsetAsync(colsum_pad, 0, (size_t)APAD * EE * sizeof(float), stream);

    {   // 1) reduce values over the sequence dimension
        dim3 grid(EE / 256, SS / 32, BB);
        colsum_kernel<<<grid, 256, 0, stream>>>(values, colsum_pad);
    }
    {   // 2) project the summed values through Wv with f32 WMMA
        dim3 grid(EE / 16);   // 64 N tiles, one wave each
        vproj_wmma_kernel<<<grid, 32, 0, stream>>>(colsum_pad, Wv, bv, hsum);
    }
    {   // 3) broadcast per-batch head vector across all S positions
        const size_t n4 = (size_t)BB * SS * (EE / 4);
        bcast_kernel<<<dim3((unsigned)(n4 / 256)), 256, 0, stream>>>(hsum, out);
    }
}